// MieScattering_27444841021729
// MI455X (gfx1250) — compile-verified
//
#include <hip/hip_runtime.h>
#include <hip/hip_bf16.h>
#include <math.h>
#include <stdint.h>

// ---------------------------------------------------------------------------
// Mie scattering, MI455X (gfx1250, wave32).
// Phase 1: deterministic two-pass mean of particle_diameter (float4 NT loads,
//          4-deep unroll -> 64B/thread of MLP -> HBM roofline).
// Phase 2: single-thread Mie coefficients a_n,b_n (n=1..50) -> 50x4 f32 table
//          [c*a_re, c*a_im, c*b_re, c*b_im]; n-dependent factors are
//          compile-time literals in the unrolled main loop.
// Phase 3: per-element 50-step recurrence run directly on q_n = P_n/sin
//          (division folded into init); packed float2 complex accumulators
//          (v_pk_fma_f32 / VOPD); coefficient table staged into LDS via
//          gfx1250 async global->LDS loads (ASYNCcnt path); NT streams.
// ---------------------------------------------------------------------------

#define N_MAX   50
#define RED_NB  256       // partial-sum blocks
#define BLK     256       // 8 waves of 32 on CDNA5

typedef float v4f __attribute__((ext_vector_type(4)));
typedef float v2f __attribute__((ext_vector_type(2)));

__global__ __launch_bounds__(BLK) void mie_reduce_d(const float* __restrict__ d,
                                                    int n,
                                                    float* __restrict__ partials) {
    __shared__ float smem[BLK];
    const int tid    = threadIdx.x;
    const int gid    = blockIdx.x * BLK + tid;
    const int stride = RED_NB * BLK;
    const int n4     = n >> 2;
    const v4f* __restrict__ d4 = (const v4f*)d;

    float a0 = 0.0f, a1 = 0.0f, a2 = 0.0f, a3 = 0.0f;
    int i = gid;
    // 4 independent 16B NT loads in flight per trip
    for (; i + 3 * stride < n4; i += 4 * stride) {
        const v4f v0 = __builtin_nontemporal_load(d4 + i);
        const v4f v1 = __builtin_nontemporal_load(d4 + i + stride);
        const v4f v2 = __builtin_nontemporal_load(d4 + i + 2 * stride);
        const v4f v3 = __builtin_nontemporal_load(d4 + i + 3 * stride);
        a0 += v0.x + v0.y + v0.z + v0.w;
        a1 += v1.x + v1.y + v1.z + v1.w;
        a2 += v2.x + v2.y + v2.z + v2.w;
        a3 += v3.x + v3.y + v3.z + v3.w;
    }
    for (; i < n4; i += stride) {
        const v4f v = __builtin_nontemporal_load(d4 + i);
        a0 += v.x + v.y + v.z + v.w;
    }
    float s = (a0 + a1) + (a2 + a3);
    for (int j = 4 * n4 + gid; j < n; j += stride)   // scalar tail
        s += d[j];

    smem[tid] = s;
    __syncthreads();
    #pragma unroll
    for (int st = BLK / 2; st > 0; st >>= 1) {
        if (tid < st) smem[tid] += smem[tid + st];
        __syncthreads();
    }
    if (tid == 0) partials[blockIdx.x] = smem[0];
}

// One thread: final sum -> x_mean -> spherical Bessel upward recurrence ->
// a_n, b_n -> packed per-n table of 4 floats: [c*a_re, c*a_im, c*b_re, c*b_im]
__global__ void mie_coeff(const float* __restrict__ partials, int nPartials,
                          float invN, float* __restrict__ table) {
    if (threadIdx.x != 0 || blockIdx.x != 0) return;
    float sum = 0.0f;
    for (int i = 0; i < nPartials; ++i) sum += partials[i];

    const float WL  = 5.5e-7f;
    const float m   = 1.31f;
    const float EPS = 1e-10f;
    const float x   = ((float)M_PI / WL) * (sum * invN);
    const float mx  = m * x;

    const float sx  = sinf(x),  cx  = cosf(x);
    const float smx = sinf(mx), cmx = cosf(mx);
    const float ix  = 1.0f / x, imx = 1.0f / mx;

    // (f_{n-1}, f_n) state, starting at n=1
    float jA = sx * ix,   jB = sx * ix * ix - cx * ix;
    float yA = -cx * ix,  yB = -cx * ix * ix - sx * ix;
    float mA = smx * imx, mB = smx * imx * imx - cmx * imx;

    for (int n = 1; n <= N_MAX; ++n) {
        const float nf = (float)n;

        const float djn = mA - (nf + 1.0f) * imx * mB;
        const float D   = djn / (mB + EPS);

        const float psi  = x * jB,  psi1 = x * jA;
        const float chiB = x * yB,  chiA = x * yA;

        const float fa = D / m + nf * ix;
        const float fb = m * D + nf * ix;

        const float numa = fa * psi - psi1;
        const float dar  = numa + EPS;
        const float dai  = fa * chiB - chiA;
        const float ia2  = 1.0f / (dar * dar + dai * dai);
        const float a_re = numa * dar * ia2;
        const float a_im = -numa * dai * ia2;

        const float numb = fb * psi - psi1;
        const float dbr  = numb + EPS;
        const float dbi  = fb * chiB - chiA;
        const float ib2  = 1.0f / (dbr * dbr + dbi * dbi);
        const float b_re = numb * dbr * ib2;
        const float b_im = -numb * dbi * ib2;

        const float coef = (2.0f * nf + 1.0f) / (nf * (nf + 1.0f));

        float* e = table + (size_t)(n - 1) * 4;
        e[0] = coef * a_re; e[1] = coef * a_im;
        e[2] = coef * b_re; e[3] = coef * b_im;

        const float c2n = (2.0f * nf + 1.0f);
        const float jC = c2n * ix  * jB - jA;
        const float yC = c2n * ix  * yB - yA;
        const float mC = c2n * imx * mB - mA;
        jA = jB; jB = jC; yA = yB; yB = yC; mA = mB; mB = mC;
    }
}

__global__ __launch_bounds__(BLK) void mie_main(const float* __restrict__ theta,
                                                const float* __restrict__ gtab,
                                                float* __restrict__ out,
                                                int n) {
    __shared__ __align__(16) float tab[N_MAX * 4];   // 800 B
    const int tid = threadIdx.x;

    // --- gfx1250 async global->LDS copy of the coefficient table (50 x 16B)
    if (tid < N_MAX) {
        unsigned long long ga =
            (unsigned long long)(uintptr_t)gtab + (unsigned long long)(tid * 16);
        unsigned int lds = (unsigned int)(uintptr_t)(&tab[0]) + (unsigned int)(tid * 16);
        asm volatile("global_load_async_to_lds_b128 %0, %1, off"
                     :: "v"(lds), "v"(ga) : "memory");
    }
    asm volatile("s_wait_asynccnt 0x0" ::: "memory");
    __syncthreads();

    const int i = blockIdx.x * BLK + tid;
    if (i >= n) return;

    const float th    = __builtin_nontemporal_load(theta + i);
    const float ct    = __cosf(th);
    const float st    = __sinf(th) + 1e-10f;
    const float inv_s = 1.0f / st;

    // q_n = P^1_n / sin(theta): same linear recurrence as P, scaled once.
    // pi_n = q_cur, P_prev/sin = q_prev.
    float qp = 0.0f;                                      // q_0
    float qc = -sqrtf(fmaxf(1.0f - ct * ct, 0.0f)) * inv_s; // q_1
    v2f S1, S2;

    // n = 1 peeled: tau_1 = cos(theta)
    {
        const v4f e  = *(const v4f*)(&tab[0]);
        const v2f ca = e.xy, cb = e.zw;
        const float pi1 = qc;
        const float tau = ct;
        S1 = ca * pi1 + cb * tau;
        S2 = ca * tau + cb * pi1;
        const float qn = 3.0f * (ct * qc);   // - 2*qp, qp == 0
        qp = qc; qc = qn;
    }

    #pragma unroll
    for (int k = 1; k < N_MAX; ++k) {
        // processing order n = k+1; n-dependent factors fold to literals
        const float nf  = (float)(k + 1);
        const float np1 = (float)(k + 2);
        const float c1  = (float)(2 * k + 3) / (float)(k + 1);  // (2n+1)/n
        const float c2  = (float)(k + 2) / (float)(k + 1);      // (n+1)/n

        const v4f e  = *(const v4f*)(&tab[k * 4]);   // single ds_load_b128
        const v2f ca = e.xy, cb = e.zw;

        const float u   = qc;              // pi_n
        const float v   = qp;              // P_prev/sin
        const float t   = ct * u;          // shared by tau and recurrence
        const float tau = nf * t - np1 * v;

        S1 += ca * u + cb * tau;           // packed complex accumulate
        S2 += ca * tau + cb * u;

        const float qn = c1 * t - c2 * v;  // q_{n+1}
        qp = qc; qc = qn;
    }

    const float SCALE = (5.5e-7f * 5.5e-7f) / (4.0f * (float)M_PI * (float)M_PI);
    const float inten =
        0.5f * (S1.x * S1.x + S1.y * S1.y + S2.x * S2.x + S2.y * S2.y) * SCALE;
    __builtin_nontemporal_store(inten, out + i);
}

extern "C" void kernel_launch(void* const* d_in, const int* in_sizes, int n_in,
                              void* d_out, int out_size, void* d_ws, size_t ws_size,
                              hipStream_t stream) {
    const float* d     = (const float*)d_in[0];   // particle_diameter
    const float* theta = (const float*)d_in[1];   // scattering_angle
    float*       out   = (float*)d_out;
    const int    n     = in_sizes[0];

    float* ws       = (float*)d_ws;
    float* partials = ws;            // RED_NB floats
    float* table    = ws + 256;      // byte offset 1024 (16B aligned), 200 floats

    mie_reduce_d<<<RED_NB, BLK, 0, stream>>>(d, n, partials);
    mie_coeff<<<1, 32, 0, stream>>>(partials, RED_NB, 1.0f / (float)n, table);
    const int blocks = (n + BLK - 1) / BLK;
    mie_main<<<blocks, BLK, 0, stream>>>(theta, table, out, n);
}